// SMBBlock_17867063951714
// MI455X (gfx1250) — compile-verified
//
#include <hip/hip_runtime.h>
#include <math.h>

// ---- problem constants -----------------------------------------------------
#define B0n   4
#define DIMn  128      // model dim C
#define DIn   256      // d_inner
#define DSn   16       // d_state
#define DTRn  8        // dt_rank
#define NSn   16       // nslices
#define Ln    9216     // L = 16*24*24
#define LNSn  576      // L / NSLICES
#define E2n   512      // 2*d_inner
#define XRn   48       // x_dbl rows padded (40 -> 48 for 16-row WMMA tiles)
#define NCHn  64       // scan chunks
#define LCn   144      // Ln / NCHn

typedef __attribute__((ext_vector_type(2))) float v2f;
typedef __attribute__((ext_vector_type(8))) float v8f;

__device__ __forceinline__ float siluf(float x)     { return x / (1.f + __expf(-x)); }
__device__ __forceinline__ float softplusf(float x) { return x > 20.f ? x : log1pf(__expf(x)); }

// branch sequence-index mapping: read source position in original xz order
template<int BR>
__device__ __forceinline__ int lsrc(int l) {
  if (BR == 0) return l;                    // forward
  if (BR == 1) return (Ln - 1) - l;         // flipped
  return (l % NSn) * LNSn + (l / NSn);      // slice-transposed
}

// ---- fp32 WMMA 16x16x4, NT-way N-blocked: D(16 x 16*NT) = A(16xK)*B(K x 16*NT)
// A row-major (lda = K stride), valid rows < Mvalid (guarded load, zero-pad).
// B in "KN" layout: B[k][n] = Bs[k*ldb + n].
// Lane mapping per CDNA5 ISA 7.12.2:
//   A: m = lane%16, k = v + 2*(lane/16)   -> contiguous float2 per lane
//   B: k = v + 2*(lane/16), n = lane%16
//   C/D: m = v + 8*(lane/16), n = lane%16
template<int NT>
__device__ __forceinline__ void wmma_kn_f32(const float* __restrict__ A, int lda,
                                            int Mvalid, int m0,
                                            const float* __restrict__ Bs, int ldb,
                                            int n0, int K, int lane, v8f* acc)
{
  const int  mrow   = lane & 15;
  const int  kh     = (lane >> 4) << 1;          // 0 or 2
  const bool mvalid = (m0 + mrow) < Mvalid;
  const float* ap = A + (size_t)(m0 + mrow) * lda + kh;
  const float* bp = Bs + (size_t)kh * ldb + n0 + mrow;
  for (int k = 0; k < K; k += 4) {
    v2f a;
    a.x = mvalid ? ap[k]     : 0.f;
    a.y = mvalid ? ap[k + 1] : 0.f;
    #pragma unroll
    for (int j = 0; j < NT; ++j) {               // reuse A across NT column tiles
      v2f b;
      b.x = bp[(size_t)k * ldb + j * 16];
      b.y = bp[(size_t)(k + 1) * ldb + j * 16];
      acc[j] = __builtin_amdgcn_wmma_f32_16x16x4_f32(false, a, false, b, (short)0,
                                                     acc[j], false, false);
    }
  }
}

// ---- K1: channel LayerNorm, x:(B,C,L) -> xn:(B,C,L) ------------------------
__global__ void k_ln(const float* __restrict__ x, const float* __restrict__ w,
                     const float* __restrict__ bia, float* __restrict__ xn)
{
  int t = blockIdx.x * blockDim.x + threadIdx.x;   // one (b,l) per thread
  int b = t / Ln, l = t % Ln;
  const float* px = x + (size_t)b * DIMn * Ln + l;
  float s = 0.f, ss = 0.f;
  for (int c = 0; c < DIMn; ++c) { float v = px[(size_t)c * Ln]; s += v; ss += v * v; }
  float mu = s * (1.f / DIMn);
  float var = ss * (1.f / DIMn) - mu * mu;
  float rstd = rsqrtf(var + 1e-5f);
  float* po = xn + (size_t)b * DIMn * Ln + l;
  for (int c = 0; c < DIMn; ++c) {
    float v = px[(size_t)c * Ln];
    po[(size_t)c * Ln] = (v - mu) * rstd * w[c] + bia[c];
  }
}

// ---- K2: in_proj GEMM: xz[b,e,l] = sum_c W[e,c]*xn[b,c,l] ------------------
__global__ void k_gemm_inproj(const float* __restrict__ W,
                              const float* __restrict__ xn,
                              float* __restrict__ xz)
{
  int gt = blockIdx.x * blockDim.x + threadIdx.x;
  int wave = gt >> 5, lane = gt & 31;
  int nt = wave % (Ln / 64);
  int mt = (wave / (Ln / 64)) % (E2n / 16);
  int b  =  wave / ((Ln / 64) * (E2n / 16));
  v8f acc[4] = {};
  wmma_kn_f32<4>(W, DIMn, E2n, mt * 16,
                 xn + (size_t)b * DIMn * Ln, Ln, nt * 64, DIMn, lane, acc);
  int n = lane & 15, mb = (lane >> 4) * 8;
  float* dst = xz + (size_t)(b * E2n + mt * 16 + mb) * Ln + nt * 64 + n;
  #pragma unroll
  for (int j = 0; j < 4; ++j)
    #pragma unroll
    for (int v = 0; v < 8; ++v) dst[(size_t)v * Ln + j * 16] = acc[j][v];
}

// ---- K3: causal depthwise conv (k=4) + SiLU over branch-permuted seq -------
template<int BR>
__global__ void k_conv(const float* __restrict__ xz, const float* __restrict__ cw,
                       const float* __restrict__ cb, float* __restrict__ xc)
{
  int t = blockIdx.x * blockDim.x + threadIdx.x;   // (b,d,l')
  int l = t % Ln;
  int d = (t / Ln) % DIn;
  int b =  t / (Ln * DIn);
  const float* src = xz + (size_t)(b * E2n + d) * Ln;  // x half of xz
  float acc = cb[d];
  #pragma unroll
  for (int j = 0; j < 4; ++j) {
    int li = l - 3 + j;
    if (li >= 0) acc += cw[d * 4 + j] * src[lsrc<BR>(li)];
  }
  xc[(size_t)(b * DIn + d) * Ln + l] = siluf(acc);
}

// ---- K4: x_proj GEMM: xdbl[b,r,l] = sum_d Wx[r,d]*xc[b,d,l] (r<40, pad 48) -
__global__ void k_gemm_xdbl(const float* __restrict__ Wx,
                            const float* __restrict__ xc,
                            float* __restrict__ xdbl)
{
  int gt = blockIdx.x * blockDim.x + threadIdx.x;
  int wave = gt >> 5, lane = gt & 31;
  int nt = wave % (Ln / 64);
  int mt = (wave / (Ln / 64)) % (XRn / 16);
  int b  =  wave / ((Ln / 64) * (XRn / 16));
  v8f acc[4] = {};
  wmma_kn_f32<4>(Wx, DIn, DTRn + 2 * DSn, mt * 16,
                 xc + (size_t)b * DIn * Ln, Ln, nt * 64, DIn, lane, acc);
  int n = lane & 15, mb = (lane >> 4) * 8;
  float* dst = xdbl + (size_t)(b * XRn + mt * 16 + mb) * Ln + nt * 64 + n;
  #pragma unroll
  for (int j = 0; j < 4; ++j)
    #pragma unroll
    for (int v = 0; v < 8; ++v) dst[(size_t)v * Ln + j * 16] = acc[j][v];
}

// ---- K5a: chunked scan pass A: per-chunk (P = prod a, S = chunk state) -----
// thread t -> (b, d, chunk, n); serial depth LCn=144 only.
__global__ void k_scan_partial(const float* __restrict__ xc,
                               const float* __restrict__ xdbl,
                               const float* __restrict__ wdt,
                               const float* __restrict__ bdt,
                               const float* __restrict__ Alog,
                               float* __restrict__ Pout, float* __restrict__ Sout)
{
  int t = blockIdx.x * blockDim.x + threadIdx.x;
  int n = t & 15;
  int c = (t >> 4) & (NCHn - 1);
  int d = (t >> 10) & (DIn - 1);
  int b =  t >> 18;

  float A_dn = -__expf(Alog[d * DSn + n]);
  float dtb  = bdt[d];
  float w[DTRn];
  #pragma unroll
  for (int r = 0; r < DTRn; ++r) w[r] = wdt[d * DTRn + r];

  const float* pr = xdbl + (size_t)(b * XRn) * Ln;
  const float* pB = xdbl + (size_t)(b * XRn + DTRn + n) * Ln;
  const float* pu = xc + (size_t)(b * DIn + d) * Ln;

  int l0 = c * LCn;
  float P = 1.f, S = 0.f;
  for (int i = 0; i < LCn; ++i) {
    int l = l0 + i;
    float s = dtb;
    #pragma unroll
    for (int r = 0; r < DTRn; ++r) s += w[r] * pr[(size_t)r * Ln + l];
    float dtv = softplusf(s);
    float a = __expf(dtv * A_dn);
    P *= a;
    S = S * a + (dtv * pu[l]) * pB[l];
  }
  int idx = ((b * DIn + d) * DSn + n) * NCHn + c;
  Pout[idx] = P;
  Sout[idx] = S;
}

// ---- K5b: carry pass: sequential combine over 64 chunks --------------------
__global__ void k_scan_carry(const float* __restrict__ P, const float* __restrict__ S,
                             float* __restrict__ H0)
{
  int t = blockIdx.x * blockDim.x + threadIdx.x;   // (b,d,n) flat, 16384
  const float* pP = P + (size_t)t * NCHn;
  const float* pS = S + (size_t)t * NCHn;
  float* pH = H0 + (size_t)t * NCHn;
  float h = 0.f;
  for (int c = 0; c < NCHn; ++c) {
    pH[c] = h;                       // state entering chunk c
    h = h * pP[c] + pS[c];
  }
}

// ---- K5c: replay pass: per-chunk scan from carried state, emit gated y -----
template<int BR>
__global__ void k_scan_final(const float* __restrict__ xz,
                             const float* __restrict__ xc,
                             const float* __restrict__ xdbl,
                             const float* __restrict__ wdt,
                             const float* __restrict__ bdt,
                             const float* __restrict__ Alog,
                             const float* __restrict__ Dvec,
                             const float* __restrict__ H0,
                             float* __restrict__ outsum)
{
  int t = blockIdx.x * blockDim.x + threadIdx.x;
  int n = t & 15;
  int c = (t >> 4) & (NCHn - 1);
  int d = (t >> 10) & (DIn - 1);
  int b =  t >> 18;

  float A_dn = -__expf(Alog[d * DSn + n]);
  float Dd   = Dvec[d];
  float dtb  = bdt[d];
  float w[DTRn];
  #pragma unroll
  for (int r = 0; r < DTRn; ++r) w[r] = wdt[d * DTRn + r];

  const float* pr = xdbl + (size_t)(b * XRn) * Ln;
  const float* pB = xdbl + (size_t)(b * XRn + DTRn + n) * Ln;
  const float* pC = xdbl + (size_t)(b * XRn + DTRn + DSn + n) * Ln;
  const float* pu = xc + (size_t)(b * DIn + d) * Ln;
  const float* pz = xz + (size_t)(b * E2n + DIn + d) * Ln;
  float* po = outsum + (size_t)(b * DIn + d) * Ln;

  float h = H0[(size_t)((b * DIn + d) * DSn + n) * NCHn + c];
  int l0 = c * LCn;
  for (int i = 0; i < LCn; ++i) {
    int l = l0 + i;
    float s = dtb;
    #pragma unroll
    for (int r = 0; r < DTRn; ++r) s += w[r] * pr[(size_t)r * Ln + l];
    float dtv = softplusf(s);
    float u = pu[l];
    h = h * __expf(dtv * A_dn) + (dtv * u) * pB[l];
    float y = h * pC[l];
    y += __shfl_xor(y, 8);
    y += __shfl_xor(y, 4);
    y += __shfl_xor(y, 2);
    y += __shfl_xor(y, 1);
    if (n == 0) {
      int ls = lsrc<BR>(l);
      float o = (y + Dd * u) * siluf(pz[ls]);
      if (BR == 0) po[ls] = o; else po[ls] += o;
    }
  }
}

// ---- K6: out_proj GEMM + residual: xm[b,c,l] = sum_d Wo[c,d]*os[b,d,l]+x ---
__global__ void k_gemm_outproj(const float* __restrict__ Wo,
                               const float* __restrict__ os,
                               const float* __restrict__ x,
                               float* __restrict__ xm)
{
  int gt = blockIdx.x * blockDim.x + threadIdx.x;
  int wave = gt >> 5, lane = gt & 31;
  int nt = wave % (Ln / 64);
  int mt = (wave / (Ln / 64)) % (DIMn / 16);
  int b  =  wave / ((Ln / 64) * (DIMn / 16));
  v8f acc[4] = {};
  wmma_kn_f32<4>(Wo, DIn, DIMn, mt * 16,
                 os + (size_t)b * DIn * Ln, Ln, nt * 64, DIn, lane, acc);
  int n = lane & 15, mb = (lane >> 4) * 8;
  size_t base = (size_t)(b * DIMn + mt * 16 + mb) * Ln + nt * 64 + n;
  #pragma unroll
  for (int j = 0; j < 4; ++j)
    #pragma unroll
    for (int v = 0; v < 8; ++v) {
      size_t idx = base + (size_t)v * Ln + j * 16;
      xm[idx] = acc[j][v] + x[idx];
    }
}

// ---- K7: final channel norm: out = xm + (nw*(xm-mu)*rstd + nb) -------------
__global__ void k_final(const float* __restrict__ xm, const float* __restrict__ nw,
                        const float* __restrict__ nb, float* __restrict__ out)
{
  int t = blockIdx.x * blockDim.x + threadIdx.x;
  int b = t / Ln, l = t % Ln;
  const float* px = xm + (size_t)b * DIMn * Ln + l;
  float s = 0.f, ss = 0.f;
  for (int c = 0; c < DIMn; ++c) { float v = px[(size_t)c * Ln]; s += v; ss += v * v; }
  float mu = s * (1.f / DIMn);
  float var = ss * (1.f / DIMn) - mu * mu;
  float rstd = rsqrtf(var + 1e-6f);
  float* po = out + (size_t)b * DIMn * Ln + l;
  for (int c = 0; c < DIMn; ++c) {
    float v = px[(size_t)c * Ln];
    po[(size_t)c * Ln] = v + (nw[c] * (v - mu) * rstd + nb[c]);
  }
}

// ---- branch driver ---------------------------------------------------------
template<int BR>
static void run_branch(const float* xz, float* xc, float* xdbl,
                       float* Pbuf, float* Sbuf, float* H0, float* outsum,
                       const float* cw, const float* cb, const float* xpw,
                       const float* dtw, const float* dtbias, const float* alog,
                       const float* dv, hipStream_t stream)
{
  k_conv<BR><<<(B0n * DIn * Ln) / 256, 256, 0, stream>>>(xz, cw, cb, xc);
  k_gemm_xdbl<<<(B0n * (XRn / 16) * (Ln / 64)) / 8, 256, 0, stream>>>(xpw, xc, xdbl);
  k_scan_partial<<<(B0n * DIn * NCHn * DSn) / 256, 256, 0, stream>>>(
      xc, xdbl, dtw, dtbias, alog, Pbuf, Sbuf);
  k_scan_carry<<<(B0n * DIn * DSn) / 256, 256, 0, stream>>>(Pbuf, Sbuf, H0);
  k_scan_final<BR><<<(B0n * DIn * NCHn * DSn) / 256, 256, 0, stream>>>(
      xz, xc, xdbl, dtw, dtbias, alog, dv, H0, outsum);
}

extern "C" void kernel_launch(void* const* d_in, const int* in_sizes, int n_in,
                              void* d_out, int out_size, void* d_ws, size_t ws_size,
                              hipStream_t stream) {
  (void)in_sizes; (void)n_in; (void)out_size; (void)ws_size;

  const float* x         = (const float*)d_in[0];
  const float* ln_w      = (const float*)d_in[1];
  const float* ln_b      = (const float*)d_in[2];
  const float* in_proj_w = (const float*)d_in[3];
  const float* out_proj_w= (const float*)d_in[4];
  // per-branch params: base 5 + br*7 : conv_w, conv_b, xproj_w, dtproj_w, dtproj_b, A_log, D
  const float* nm_w = (const float*)d_in[26];
  const float* nm_b = (const float*)d_in[27];

  float* ws     = (float*)d_ws;
  float* xn     = ws;
  float* xz     = xn     + (size_t)B0n * DIMn * Ln;
  float* xc     = xz     + (size_t)B0n * E2n  * Ln;
  float* xdbl   = xc     + (size_t)B0n * DIn  * Ln;
  float* outsum = xdbl   + (size_t)B0n * XRn  * Ln;
  float* xmamba = outsum + (size_t)B0n * DIn  * Ln;
  float* Pbuf   = xmamba + (size_t)B0n * DIMn * Ln;
  float* Sbuf   = Pbuf   + (size_t)B0n * DIn * DSn * NCHn;
  float* H0     = Sbuf   + (size_t)B0n * DIn * DSn * NCHn;

  k_ln<<<(B0n * Ln) / 256, 256, 0, stream>>>(x, ln_w, ln_b, xn);
  k_gemm_inproj<<<(B0n * (E2n / 16) * (Ln / 64)) / 8, 256, 0, stream>>>(in_proj_w, xn, xz);

  #define BRP(i) (const float*)d_in[5 + (i)]
  run_branch<0>(xz, xc, xdbl, Pbuf, Sbuf, H0, outsum,
                BRP(0), BRP(1), BRP(2), BRP(3), BRP(4), BRP(5), BRP(6), stream);
  run_branch<1>(xz, xc, xdbl, Pbuf, Sbuf, H0, outsum,
                BRP(7), BRP(8), BRP(9), BRP(10), BRP(11), BRP(12), BRP(13), stream);
  run_branch<2>(xz, xc, xdbl, Pbuf, Sbuf, H0, outsum,
                BRP(14), BRP(15), BRP(16), BRP(17), BRP(18), BRP(19), BRP(20), stream);
  #undef BRP

  k_gemm_outproj<<<(B0n * (DIMn / 16) * (Ln / 64)) / 8, 256, 0, stream>>>(
      out_proj_w, outsum, x, xmamba);
  k_final<<<(B0n * Ln) / 256, 256, 0, stream>>>(xmamba, nm_w, nm_b, (float*)d_out);
}